// IndexedConv3D_22084721836464
// MI455X (gfx1250) — compile-verified
//
#include <hip/hip_runtime.h>

// Problem constants (from reference)
#define B_    32
#define L_    1855
#define K_    7
#define T_    40
#define CIN   8
#define COUT  32
#define TKS   5
#define TOUT  36              // T - TKS + 1

#define RB        16          // b-rows per block (one WMMA M tile)
#define NF_STRIDE 2244        // words/b-row: 7*320 + 4 pad (==4 mod 64 -> conflict-free; ==0 mod 4 -> 16B aligned)
#define KB_WORDS  (2*70*32*2) // 8960 floats: kernel in B-fragment layout [nt][c][lane][2]
#define NF_WORDS  (RB*NF_STRIDE)
#define SMEM_BYTES ((KB_WORDS + NF_WORDS) * 4)   // 179,456 B < 320 KB

typedef __attribute__((ext_vector_type(2))) float v2f;
typedef __attribute__((ext_vector_type(4))) float v4f;
typedef __attribute__((ext_vector_type(8))) float v8f;

// Types matching __builtin_amdgcn_global_load_async_to_lds_b128's params
// (diagnostic showed: "__attribute__((__vector_size__(4 * sizeof(int)))) int __device__ *")
typedef int v4i_gcc __attribute__((vector_size(16)));
typedef __attribute__((address_space(1))) v4i_gcc g_v4i;   // global
typedef __attribute__((address_space(3))) v4i_gcc l_v4i;   // LDS

__global__ __launch_bounds__(256) void idxconv_wmma(
    const float* __restrict__ x, const int* __restrict__ nbr,
    const float* __restrict__ kern, const float* __restrict__ bias,
    float* __restrict__ out)
{
    extern __shared__ float smem[];
    float* kb = smem;               // [2][70][32][2]  B fragments
    float* nf = smem + KB_WORDS;    // [16][NF_STRIDE] gathered A rows

    const int l   = blockIdx.x;
    const int b0  = blockIdx.y * RB;
    const int tid = threadIdx.x;

    // Neighbor indices for this l (uniform -> scalar loads / scalar branches)
    int nk[K_];
    #pragma unroll
    for (int k = 0; k < K_; ++k) nk[k] = nbr[l * K_ + k];

    // ---- Stage kernel into LDS in exact WMMA B-fragment layout ----
    for (int d = tid; d < KB_WORDS; d += 256) {
        int nt   = d / 4480;
        int rem  = d % 4480;
        int c    = rem / 64;
        int lp   = rem % 64;
        int lane = lp >> 1, j = lp & 1;
        int k    = c / 10, m = c % 10;
        int tau  = m >> 1, i0 = (m & 1) * 4;
        int i    = i0 + ((lane >> 4) << 1) + j;
        int o    = nt * 16 + (lane & 15);
        kb[d] = kern[((k * TKS + tau) * CIN + i) * COUT + o];
    }

    // ---- Stage gathered neighbor features into LDS ----
    // per (b,k): 320 contiguous floats of x[b0+b, nk, :, :] = 80 float4
    for (int q = tid; q < RB * K_ * 80; q += 256) {
        int b   = q / (K_ * 80);
        int rem = q % (K_ * 80);
        int k   = rem / 80;
        int v   = rem % 80;
        int nl  = nk[k];
        if (nl >= 0) {
            const float* src = x + ((size_t)(b0 + b) * L_ + nl) * (T_ * CIN) + v * 4;
            float*       dst = nf + b * NF_STRIDE + k * 320 + v * 4;
#if __has_builtin(__builtin_amdgcn_global_load_async_to_lds_b128)
            // Direct memory->LDS async copy (ASYNCcnt), no VGPR round-trip.
            __builtin_amdgcn_global_load_async_to_lds_b128(
                (g_v4i*)src, (l_v4i*)dst, /*offset=*/0, /*cpol=*/0);
#else
            *(v4f*)dst = *(const v4f*)src;
#endif
        }
    }
#if __has_builtin(__builtin_amdgcn_global_load_async_to_lds_b128)
#if __has_builtin(__builtin_amdgcn_s_wait_asynccnt)
    __builtin_amdgcn_s_wait_asynccnt(0);
#else
    asm volatile("s_wait_asynccnt 0x0" ::: "memory");
#endif
#endif
    __syncthreads();

    const int wid  = tid >> 5;
    const int lane = tid & 31;
    const int nt   = wid & 1;        // which 16-wide C_OUT tile
    const int tb   = wid >> 1;       // t phase 0..3 ; t = tb + 4*it
    const int n    = lane & 15;
    const int hi   = lane >> 4;
    const float bo = bias[nt * 16 + n];

    // A fragment: 16x4 f32 layout: lane(0-15)=row M=n, K={0,1}; lane+16: K={2,3}
    const float* aBase = nf + n * NF_STRIDE + 2 * hi + tb * 8;
    const v2f*   bBase = (const v2f*)kb + (nt * 70) * 32 + lane;

    // 9 independent accumulators (one per t) -> 9-way ILP on the matrix pipe
    v8f acc[9];
    #pragma unroll
    for (int it = 0; it < 9; ++it)
        acc[it] = (v8f){ bo, bo, bo, bo, bo, bo, bo, bo };

    #pragma unroll 1
    for (int k = 0; k < K_; ++k) {
        if (nk[k] < 0) continue;          // uniform skip == zeroed A columns

        // Hoist the 10 B fragments of this k into registers (reused by 9 t's)
        v2f bfrag[10];
        #pragma unroll
        for (int m = 0; m < 10; ++m)
            bfrag[m] = bBase[(k * 10 + m) * 32];

        const float* ak = aBase + k * 320;
        #pragma unroll
        for (int m = 0; m < 10; ++m) {
            int tau = m >> 1, i0 = (m & 1) * 4;
            #pragma unroll
            for (int it = 0; it < 9; ++it) {
                v2f a = *(const v2f*)(ak + tau * 8 + i0 + it * 32);
                acc[it] = __builtin_amdgcn_wmma_f32_16x16x4_f32(
                            false, a, false, bfrag[m],
                            (short)0, acc[it], false, false);
            }
        }
    }

    // C/D layout: VGPR r -> M = r + 8*hi (b index), col = nt*16 + n
    const size_t obase = ((size_t)(b0 + 8 * hi) * L_ + l) * (TOUT * COUT)
                         + (size_t)tb * COUT + nt * 16 + n;
    #pragma unroll
    for (int it = 0; it < 9; ++it) {
        #pragma unroll
        for (int r = 0; r < 8; ++r) {
            size_t off = obase + (size_t)r * ((size_t)L_ * TOUT * COUT)
                       + (size_t)it * 4 * COUT;
            out[off] = acc[it][r];
        }
    }
}

extern "C" void kernel_launch(void* const* d_in, const int* in_sizes, int n_in,
                              void* d_out, int out_size, void* d_ws, size_t ws_size,
                              hipStream_t stream) {
    const float* x    = (const float*)d_in[0];
    const int*   nbr  = (const int*)d_in[1];
    const float* kern = (const float*)d_in[2];
    const float* bias = (const float*)d_in[3];
    float*       out  = (float*)d_out;
    (void)in_sizes; (void)n_in; (void)out_size; (void)d_ws; (void)ws_size;

    (void)hipFuncSetAttribute((const void*)idxconv_wmma,
                              hipFuncAttributeMaxDynamicSharedMemorySize, SMEM_BYTES);

    dim3 grid(L_, B_ / RB);
    idxconv_wmma<<<grid, 256, SMEM_BYTES, stream>>>(x, nbr, kern, bias, out);
}